// GCN_77584289235636
// MI455X (gfx1250) — compile-verified
//
#include <hip/hip_runtime.h>

// ---------------------------------------------------------------------------
// GCN 2-layer forward for MI455X (gfx1250), wave32 + WMMA f32 16x16x4.
//   deg -> inv_sqrt -> scatter(agg1) -> WMMA gemm1(+relu, W in LDS)
//   -> scatter(agg2) -> WMMA gemm2 -> out
// ---------------------------------------------------------------------------

typedef __attribute__((ext_vector_type(2))) float v2f;
typedef __attribute__((ext_vector_type(8))) float v8f;

#define IN_DIM  128
#define HID_DIM 128
#define OUT_DIM 64

// -------------------------- degree histogram -------------------------------
__global__ void gcn_degree_kernel(const int* __restrict__ src,
                                  const int* __restrict__ dst,
                                  int* __restrict__ deg_out,
                                  int* __restrict__ deg_in, int E) {
    int e = blockIdx.x * blockDim.x + threadIdx.x;
    if (e < E) {
        atomicAdd(&deg_out[src[e]], 1);
        atomicAdd(&deg_in[dst[e]], 1);
    }
}

// inv[i] = rsqrt(max(deg,1)) over 2N contiguous entries (out then in)
__global__ void gcn_invsqrt_kernel(const int* __restrict__ deg,
                                   float* __restrict__ inv, int n2) {
    int i = blockIdx.x * blockDim.x + threadIdx.x;
    if (i < n2) {
        float d = (float)deg[i];
        inv[i] = rsqrtf(fmaxf(d, 1.0f));
    }
}

// ---------------- edge scatter: agg[dst] += x[src] * inv_out[src] ----------
// One wave (32 lanes) per edge; each lane moves 4 contiguous floats (float4).
// Feature matrix + accumulation target both fit in the 192MB L2, so these
// random-access atomics are L2-resident.
__global__ __launch_bounds__(256)
void gcn_scatter_kernel(const float* __restrict__ x,
                        const int* __restrict__ src,
                        const int* __restrict__ dst,
                        const float* __restrict__ inv_out,
                        float* __restrict__ agg, int E) {
    long long t = (long long)blockIdx.x * blockDim.x + threadIdx.x;
    int e = (int)(t >> 5);
    if (e >= E) return;
    int lane = (int)(t & 31);
    int s = src[e];
    int d = dst[e];
    float sc = inv_out[s];
    const float4* xp = (const float4*)(x + (size_t)s * 128) + lane;
    float4 v = *xp;
    float* ap = agg + (size_t)d * 128 + lane * 4;
    unsafeAtomicAdd(ap + 0, v.x * sc);
    unsafeAtomicAdd(ap + 1, v.y * sc);
    unsafeAtomicAdd(ap + 2, v.z * sc);
    unsafeAtomicAdd(ap + 3, v.w * sc);
}

// ---------------- fused GEMM: out = act(diag(inv_in)*A @ W + b) ------------
// A: [N, K] (agg), W: [K, NOUT] row-major, out: [N, NOUT].
// 256 threads = 8 waves; block covers 128 rows (16 per wave). W is staged
// once per block into LDS in pair-interleaved form:
//     lds_w[(k/2)*S + n*2 + (k&1)],  S = 2*NOUT + 32 dwords
// so a lane's B fragment (W[ka][n], W[ka+1][n]) is one contiguous 8B
// ds_load_b64 into an even-aligned VGPR pair. S = 32 (mod 64) banks makes
// the two half-waves (pair-rows j and j+1) hit disjoint 32-bank halves.
// Each wave keeps NT v8f accumulators and reads its A rows exactly once.
//
// Fragment layouts (ISA 7.12.2, wave32):
//   A 16x4 : lane L -> row L%16;  VGPR0 = K = 2*(L/16), VGPR1 = K = 2*(L/16)+1
//   B 4x16 : lane L -> col L%16;  VGPR0 = K = 2*(L/16), VGPR1 = K = 2*(L/16)+1
//   C 16x16: lane L -> col L%16;  VGPR r = row r + 8*(L/16)
template <int K, int NOUT, bool RELU>
__global__ __launch_bounds__(256)
void gcn_wmma_gemm_kernel(const float* __restrict__ A,
                          const float* __restrict__ W,
                          const float* __restrict__ bias,
                          const float* __restrict__ inv_in,
                          float* __restrict__ out, int N) {
    constexpr int S  = 2 * NOUT + 32;         // pair-row stride (dwords)
    constexpr int NT = NOUT / 16;             // number of 16-col tiles
    __shared__ float lds_w[(K / 2) * S];

    // ---- cooperative load of W into LDS, pair-interleaved ----
    for (int idx = threadIdx.x; idx < K * NOUT; idx += blockDim.x) {
        int k = idx / NOUT;
        int n = idx - k * NOUT;
        lds_w[(k >> 1) * S + n * 2 + (k & 1)] = W[idx];
    }
    __syncthreads();

    const int lane = threadIdx.x & 31;
    const int wave = threadIdx.x >> 5;
    const int g = lane >> 4;                  // half-wave id (0/1)
    const int mr = lane & 15;                 // row-in-tile (A) / col (B, C)

    const int row0 = blockIdx.x * 128 + wave * 16;   // this wave's row base
    int row = row0 + mr;
    if (row >= N) row = N - 1;                // clamp: EXEC stays all-1s
    const float s = inv_in[row];
    const float* __restrict__ Arow = A + (size_t)row * K;

    v8f acc[NT];
#pragma unroll
    for (int t = 0; t < NT; ++t) acc[t] = (v8f){};

#pragma unroll 4
    for (int k0 = 0; k0 < K; k0 += 4) {
        const int ka = k0 + 2 * g;            // this half-wave's K base
        v2f a = *(const v2f*)(Arow + ka);     // contiguous, 8B aligned
        a.x *= s;
        a.y *= s;
        // pair-row for (ka, ka+1) is ka/2 = k0/2 + g
        const v2f* __restrict__ wp =
            (const v2f*)(lds_w + (size_t)(k0 / 2 + g) * S);
#pragma unroll
        for (int t = 0; t < NT; ++t) {
            v2f b = wp[t * 16 + mr];          // one ds_load_b64, even pair
            acc[t] = __builtin_amdgcn_wmma_f32_16x16x4_f32(
                /*neg_a=*/false, a, /*neg_b=*/false, b,
                /*c_mod=*/(short)0, acc[t], /*reuse_a=*/false, /*reuse_b=*/false);
        }
    }

#pragma unroll
    for (int t = 0; t < NT; ++t) {
        const int Nbase = t * 16;
        const float bv = bias[Nbase + mr];
#pragma unroll
        for (int r = 0; r < 8; ++r) {
            const int orow = row0 + r + 8 * g;
            if (orow < N) {
                float v = acc[t][r] + bv;
                if (RELU) v = fmaxf(v, 0.0f);
                out[(size_t)orow * NOUT + Nbase + mr] = v;
            }
        }
    }
}

// ---------------------------------------------------------------------------
extern "C" void kernel_launch(void* const* d_in, const int* in_sizes, int n_in,
                              void* d_out, int out_size, void* d_ws, size_t ws_size,
                              hipStream_t stream) {
    const float* in_feat = (const float*)d_in[0];
    const float* W1      = (const float*)d_in[1];
    const float* b1      = (const float*)d_in[2];
    const float* W2      = (const float*)d_in[3];
    const float* b2      = (const float*)d_in[4];
    const int*   src     = (const int*)d_in[5];
    const int*   dst     = (const int*)d_in[6];
    float*       out     = (float*)d_out;

    const int N = in_sizes[0] / IN_DIM;   // 100000
    const int E = in_sizes[5];            // 1600000

    // workspace layout
    char* ws = (char*)d_ws;
    int*   deg   = (int*)ws;                               // 2N ints (out|in)
    float* inv   = (float*)(ws + (size_t)2 * N * 4);       // 2N floats (out|in)
    float* agg   = (float*)(ws + (size_t)4 * N * 4);       // N*128 floats
    float* hbuf  = agg + (size_t)N * HID_DIM;              // N*128 floats
    float* inv_out = inv;
    float* inv_in  = inv + N;

    // 1) zero degree counters and layer-1 accumulation buffer
    hipMemsetAsync(deg, 0, (size_t)2 * N * sizeof(int), stream);
    hipMemsetAsync(agg, 0, (size_t)N * HID_DIM * sizeof(float), stream);

    // 2) degrees + inverse-sqrt normalization
    {
        int blk = 256;
        gcn_degree_kernel<<<(E + blk - 1) / blk, blk, 0, stream>>>(
            src, dst, deg, deg + N, E);
        gcn_invsqrt_kernel<<<(2 * N + blk - 1) / blk, blk, 0, stream>>>(
            deg, inv, 2 * N);
    }

    // 3) layer-1 edge aggregation: agg[dst] += in_feat[src] * inv_out[src]
    {
        long long threads = (long long)E * 32;
        int blk = 256;
        long long grid = (threads + blk - 1) / blk;
        gcn_scatter_kernel<<<(unsigned)grid, blk, 0, stream>>>(
            in_feat, src, dst, inv_out, agg, E);
    }

    // 4) h = relu(diag(inv_in) * agg @ W1 + b1)   [N, 128]
    {
        int grid = (N + 127) / 128;
        gcn_wmma_gemm_kernel<IN_DIM, HID_DIM, true>
            <<<grid, 256, 0, stream>>>(agg, W1, b1, inv_in, hbuf, N);
    }

    // 5) layer-2 aggregation into the (re-zeroed) agg buffer
    hipMemsetAsync(agg, 0, (size_t)N * HID_DIM * sizeof(float), stream);
    {
        long long threads = (long long)E * 32;
        int blk = 256;
        long long grid = (threads + blk - 1) / blk;
        gcn_scatter_kernel<<<(unsigned)grid, blk, 0, stream>>>(
            hbuf, src, dst, inv_out, agg, E);
    }

    // 6) out = diag(inv_in) * agg @ W2 + b2   [N, 64]
    {
        int grid = (N + 127) / 128;
        gcn_wmma_gemm_kernel<HID_DIM, OUT_DIM, false>
            <<<grid, 256, 0, stream>>>(agg, W2, b2, inv_in, out, N);
    }
}